// DSSLayer_22153441313127
// MI455X (gfx1250) — compile-verified
//
#include <hip/hip_runtime.h>
#include <math.h>

// Problem constants (match reference)
#define BB 8
#define HH 128
#define LL 4096
#define NN 64
#define CHUNK 64
#define NCH (LL / CHUNK)        // 64 chunks
#define COLS (BB * NCH)         // 512 matmul columns (b*64 + chunk)
#define EPSV 1e-7f

typedef float v2f __attribute__((ext_vector_type(2)));
typedef float v8f __attribute__((ext_vector_type(8)));

// ---------------------------------------------------------------------------
// Kernel 1: per-head coefficient build.
//   z = exp(dt*Lambda); s = (1 - z^L)/(1 - z); c = (W/Lambda)*conj(s)/(|s|^2+eps)
//   Emits: Zr/Zi[h][n][s] = Re/Im(z^(C-1-s))        (state input projection)
//          CR[h][t][n]    = Re(c*z^(t+1)), CIn = -Im (output projection, CI pre-negated)
//          T[h][t][s]     = K[t-s] (t>=s)           (intra-chunk Toeplitz)
//          zC[h][n]       = z^C                     (scan carry)
// One block per head h, 64 threads (thread = state index n).
// ---------------------------------------------------------------------------
__global__ void __launch_bounds__(64)
dss_coeffs(const float* __restrict__ Lre, const float* __restrict__ Lim,
           const float* __restrict__ Wri, const float* __restrict__ logstep,
           float* __restrict__ Zr, float* __restrict__ Zi,
           float* __restrict__ CR, float* __restrict__ CIn,
           float* __restrict__ T, float* __restrict__ zCr, float* __restrict__ zCi) {
    __shared__ float sdtr[NN], sdti[NN], scr[NN], sci[NN], sK[CHUNK];
    const int h = blockIdx.x;
    const int n = threadIdx.x;

    const float lre = Lre[h * NN + n], lim = Lim[h * NN + n];
    const float dt  = __expf(logstep[h]);
    const float ar  = dt * lre, ai = dt * lim;         // log z (Re(ar) < 0)

    // z
    float cz, sz; sincosf(ai, &sz, &cz);
    const float er = expf(ar);
    const float zr = er * cz, zi = er * sz;
    // z^L (|z|<1 strictly -> expf may underflow to 0, fine)
    float cL, sL; sincosf(ai * (float)LL, &sL, &cL);
    const float eL = expf(ar * (float)LL);
    const float zLr = eL * cL, zLi = eL * sL;
    // s = (1 - z^L) / (1 - z)
    const float nr = 1.f - zLr, ni = -zLi;
    const float dr = 1.f - zr,  di = -zi;
    const float dd = dr * dr + di * di;
    const float sr = (nr * dr + ni * di) / dd;
    const float si = (ni * dr - nr * di) / dd;
    // b = W / Lambda
    const float wr = Wri[(h * NN + n) * 2 + 0], wi = Wri[(h * NN + n) * 2 + 1];
    const float ll = lre * lre + lim * lim;
    const float br = (wr * lre + wi * lim) / ll;
    const float bi = (wi * lre - wr * lim) / ll;
    // c = b * conj(s) / (|s|^2 + eps)
    const float ss = sr * sr + si * si + EPSV;
    const float cr = (br * sr + bi * si) / ss;
    const float ci = (bi * sr - br * si) / ss;

    // z^C carry
    float cC, sC; sincosf(ai * (float)CHUNK, &sC, &cC);
    const float eC = expf(ar * (float)CHUNK);
    zCr[h * NN + n] = eC * cC;
    zCi[h * NN + n] = eC * sC;

    // Z rows: Z[n][s] = z^(C-1-s)
    {
        float pr = 1.f, pi_ = 0.f;
        float* zrrow = Zr + (h * NN + n) * CHUNK;
        float* zirow = Zi + (h * NN + n) * CHUNK;
        for (int k = 0; k < CHUNK; ++k) {
            zrrow[CHUNK - 1 - k] = pr;
            zirow[CHUNK - 1 - k] = pi_;
            const float tr = pr * zr - pi_ * zi;
            pi_ = pr * zi + pi_ * zr;
            pr = tr;
        }
    }
    // C columns: CR[t][n] = Re(c*z^(t+1)); CIn[t][n] = -Im(c*z^(t+1))
    {
        float qr = cr * zr - ci * zi;
        float qi = cr * zi + ci * zr;
        for (int t = 0; t < CHUNK; ++t) {
            CR [(h * CHUNK + t) * NN + n] =  qr;
            CIn[(h * CHUNK + t) * NN + n] = -qi;
            const float tr = qr * zr - qi * zi;
            qi = qr * zi + qi * zr;
            qr = tr;
        }
    }

    sdtr[n] = ar; sdti[n] = ai; scr[n] = cr; sci[n] = ci;
    __syncthreads();

    // K[t] = Re(sum_n c_n * z_n^t), t = thread id
    {
        const float t = (float)n;
        float acc = 0.f;
        for (int j = 0; j < NN; ++j) {
            const float e = expf(sdtr[j] * t);
            float cs, sn; sincosf(sdti[j] * t, &sn, &cs);
            acc += e * (scr[j] * cs - sci[j] * sn);
        }
        sK[n] = acc;
    }
    __syncthreads();

    // T[t][s] = K[t-s] lower-triangular Toeplitz
    for (int idx = n; idx < CHUNK * CHUNK; idx += NN) {
        const int t = idx >> 6, s = idx & 63;
        T[(h * CHUNK + t) * CHUNK + s] = (t >= s) ? sK[t - s] : 0.f;
    }
}

// ---------------------------------------------------------------------------
// Kernel 2: state input projection via fp32 WMMA.
//   P[h][col][n] = sum_s Z[h][n][s] * Uch[s][col]      (real & imag)
// where Uch[s][col=b*64+ch] = u[b][h][ch*64+s] (contiguous along s).
// P is stored n-innermost so this kernel's epilogue is b128 stores and
// kernel 4's B-operand reads are contiguous b64 loads.
// One wave per 16x16 output tile; 16 K-steps x 2 WMMA (shared B operand).
// ---------------------------------------------------------------------------
__global__ void __launch_bounds__(256)
dss_proj(const float* __restrict__ u,
         const float* __restrict__ Zr, const float* __restrict__ Zi,
         float* __restrict__ Pr, float* __restrict__ Pi) {
    const int lane  = threadIdx.x & 31;
    const int wave  = threadIdx.x >> 5;
    const int tile  = blockIdx.x * 8 + wave;        // H * 4 * 32 tiles
    const int h     = tile >> 7;
    const int rem   = tile & 127;
    const int rt    = rem >> 5;                      // row tile (n)
    const int ct    = rem & 31;                      // col tile
    const int m     = lane & 15;
    const int khalf = lane >> 4;
    const int k0    = khalf * 2;

    const int arow = rt * 16 + m;                    // A row (state n)
    const int col  = ct * 16 + m;                    // B col (b,chunk)
    const float* __restrict__ ucol =
        u + ((size_t)(col >> 6) * HH + h) * LL + (size_t)(col & 63) * CHUNK;
    const float* __restrict__ zrrow = Zr + (size_t)(h * NN + arow) * CHUNK;
    const float* __restrict__ zirow = Zi + (size_t)(h * NN + arow) * CHUNK;

    v8f accr = {}, acci = {};
#pragma unroll
    for (int kk = 0; kk < CHUNK; kk += 4) {
        const v2f aR = *(const v2f*)(zrrow + kk + k0);
        const v2f aI = *(const v2f*)(zirow + kk + k0);
        const v2f b  = *(const v2f*)(ucol + kk + k0);
        accr = __builtin_amdgcn_wmma_f32_16x16x4_f32(false, aR, false, b, (short)0, accr, false, false);
        acci = __builtin_amdgcn_wmma_f32_16x16x4_f32(false, aI, false, b, (short)0, acci, false, false);
    }

    // D rows nbase..nbase+7 are contiguous n indices for this lane's column.
    const int nbase = rt * 16 + 8 * khalf;
    const size_t o = ((size_t)(h * COLS + col)) * NN + nbase;   // 32B aligned
    float4 r0, r1, i0, i1;
    r0.x = accr[0]; r0.y = accr[1]; r0.z = accr[2]; r0.w = accr[3];
    r1.x = accr[4]; r1.y = accr[5]; r1.z = accr[6]; r1.w = accr[7];
    i0.x = acci[0]; i0.y = acci[1]; i0.z = acci[2]; i0.w = acci[3];
    i1.x = acci[4]; i1.y = acci[5]; i1.z = acci[6]; i1.w = acci[7];
    *(float4*)(Pr + o)     = r0;
    *(float4*)(Pr + o + 4) = r1;
    *(float4*)(Pi + o)     = i0;
    *(float4*)(Pi + o + 4) = i1;
}

// ---------------------------------------------------------------------------
// Kernel 3: sequential carry scan over chunks (in place).
//   x(i) = z^C * x(i-1) + P(col=(b,i)); P(col) <- x(i)   (end-of-chunk states)
// One thread per (h, n, b), n fastest -> each wave touches 32 consecutive
// floats per step. 65536 threads, 64 serial steps.
// ---------------------------------------------------------------------------
__global__ void __launch_bounds__(256)
dss_scan(float* __restrict__ Pr, float* __restrict__ Pi,
         const float* __restrict__ zCr, const float* __restrict__ zCi) {
    const int tid = blockIdx.x * blockDim.x + threadIdx.x;   // H*B*N
    const int n = tid & 63;
    const int b = (tid >> 6) & 7;
    const int h = tid >> 9;
    const size_t base = ((size_t)(h * COLS + b * NCH)) * NN + n;
    const float zr = zCr[h * NN + n], zi = zCi[h * NN + n];
    float xr = 0.f, xi = 0.f;
    for (int i = 0; i < NCH; ++i) {
        const size_t o = base + (size_t)i * NN;
        const float pr = Pr[o], pi_ = Pi[o];
        const float nxr = zr * xr - zi * xi + pr;
        const float nxi = zr * xi + zi * xr + pi_;
        Pr[o] = nxr; Pi[o] = nxi;
        xr = nxr; xi = nxi;
    }
}

// ---------------------------------------------------------------------------
// Kernel 4: fused output via fp32 WMMA.
//   Y[h][C x COLS] = T[h] @ Uch + CR[h] @ Xin_r + CIn[h] @ Xin_i
// where Xin(:, (b,ch)) = scanned state of chunk ch-1 (zero for ch==0).
// With P stored n-innermost, the X-operand k-pairs are contiguous b64 loads.
// Epilogue: y += D[h]*u, stored with two b128 stores per lane.
// ---------------------------------------------------------------------------
__global__ void __launch_bounds__(256)
dss_out(const float* __restrict__ u, const float* __restrict__ D,
        const float* __restrict__ T, const float* __restrict__ CR,
        const float* __restrict__ CIn,
        const float* __restrict__ Pr, const float* __restrict__ Pi,
        float* __restrict__ out) {
    const int lane  = threadIdx.x & 31;
    const int wave  = threadIdx.x >> 5;
    const int tile  = blockIdx.x * 8 + wave;
    const int h     = tile >> 7;
    const int rem   = tile & 127;
    const int rt    = rem >> 5;                      // row tile (t within chunk)
    const int ct    = rem & 31;                      // col tile
    const int m     = lane & 15;
    const int khalf = lane >> 4;
    const int k0    = khalf * 2;

    const int arow = rt * 16 + m;
    const int col  = ct * 16 + m;
    const int b    = col >> 6;
    const int ch   = col & 63;
    const bool first = (ch == 0);

    const float* __restrict__ ucol  = u + ((size_t)b * HH + h) * LL + (size_t)ch * CHUNK;
    const float* __restrict__ trow  = T   + (size_t)(h * CHUNK + arow) * CHUNK;
    const float* __restrict__ crrow = CR  + (size_t)(h * CHUNK + arow) * NN;
    const float* __restrict__ cirow = CIn + (size_t)(h * CHUNK + arow) * NN;
    // States entering this lane's chunk: column (b, ch-1) of the scanned P.
    const float* __restrict__ xrcol = Pr + ((size_t)(h * COLS + col - 1)) * NN;
    const float* __restrict__ xicol = Pi + ((size_t)(h * COLS + col - 1)) * NN;

    v8f acc = {};
    // intra-chunk Toeplitz
#pragma unroll
    for (int kk = 0; kk < CHUNK; kk += 4) {
        const v2f a  = *(const v2f*)(trow + kk + k0);
        const v2f bu = *(const v2f*)(ucol + kk + k0);
        acc = __builtin_amdgcn_wmma_f32_16x16x4_f32(false, a, false, bu, (short)0, acc, false, false);
    }
    // inter-chunk: real part of C @ x_in (CIn pre-negated -> plain accumulate)
#pragma unroll
    for (int kk = 0; kk < CHUNK; kk += 4) {
        const v2f aR = *(const v2f*)(crrow + kk + k0);
        const v2f aI = *(const v2f*)(cirow + kk + k0);
        v2f bxr = {0.f, 0.f}, bxi = {0.f, 0.f};
        if (!first) {
            bxr = *(const v2f*)(xrcol + kk + k0);
            bxi = *(const v2f*)(xicol + kk + k0);
        }
        acc = __builtin_amdgcn_wmma_f32_16x16x4_f32(false, aR, false, bxr, (short)0, acc, false, false);
        acc = __builtin_amdgcn_wmma_f32_16x16x4_f32(false, aI, false, bxi, (short)0, acc, false, false);
    }

    // Epilogue: y = acc + D[h]*u ; rows rt*16+8*khalf .. +7 are contiguous in t.
    const float Dh = D[h];
    const size_t obase = ((size_t)b * HH + h) * LL + (size_t)ch * CHUNK + rt * 16 + 8 * khalf;
    const float4 u0 = *(const float4*)(u + obase);
    const float4 u1 = *(const float4*)(u + obase + 4);
    float4 o0, o1;
    o0.x = acc[0] + Dh * u0.x; o0.y = acc[1] + Dh * u0.y;
    o0.z = acc[2] + Dh * u0.z; o0.w = acc[3] + Dh * u0.w;
    o1.x = acc[4] + Dh * u1.x; o1.y = acc[5] + Dh * u1.y;
    o1.z = acc[6] + Dh * u1.z; o1.w = acc[7] + Dh * u1.w;
    *(float4*)(out + obase)     = o0;
    *(float4*)(out + obase + 4) = o1;
}

// ---------------------------------------------------------------------------
extern "C" void kernel_launch(void* const* d_in, const int* in_sizes, int n_in,
                              void* d_out, int out_size, void* d_ws, size_t ws_size,
                              hipStream_t stream) {
    const float* u       = (const float*)d_in[0];
    const float* Lre     = (const float*)d_in[1];
    const float* Lim     = (const float*)d_in[2];
    const float* Wri     = (const float*)d_in[3];
    const float* Dp      = (const float*)d_in[4];
    const float* logstep = (const float*)d_in[5];
    float* out = (float*)d_out;

    // Workspace layout (floats)
    float* ws  = (float*)d_ws;
    float* Zr  = ws;                                  // H*N*C
    float* Zi  = Zr  + (size_t)HH * NN * CHUNK;
    float* CRm = Zi  + (size_t)HH * NN * CHUNK;       // H*C*N
    float* CIn = CRm + (size_t)HH * CHUNK * NN;
    float* Tm  = CIn + (size_t)HH * CHUNK * NN;       // H*C*C
    float* zCr = Tm  + (size_t)HH * CHUNK * CHUNK;    // H*N
    float* zCi = zCr + (size_t)HH * NN;
    float* Pr  = zCi + (size_t)HH * NN;               // H*COLS*N
    float* Pi  = Pr  + (size_t)HH * COLS * NN;        // total ~44 MB

    dss_coeffs<<<HH, NN, 0, stream>>>(Lre, Lim, Wri, logstep,
                                      Zr, Zi, CRm, CIn, Tm, zCr, zCi);

    const int tiles = HH * (NN / 16) * (COLS / 16);   // 16384 waves
    dss_proj<<<tiles / 8, 256, 0, stream>>>(u, Zr, Zi, Pr, Pi);

    dss_scan<<<(HH * NN * BB) / 256, 256, 0, stream>>>(Pr, Pi, zCr, zCi);

    dss_out<<<tiles / 8, 256, 0, stream>>>(u, Dp, Tm, CRm, CIn, Pr, Pi, out);
}